// BEVTransform_90108413870310
// MI455X (gfx1250) — compile-verified
//
#include <hip/hip_runtime.h>
#include <math.h>

// BEV bilinear resample for MI455X (gfx1250, wave32).
//   x: (8, 64, 192, 320) f32;  out tuple = (out[B,C,H,W], scales[B], centers[B,3])
// Memory-bound: ~252 MB HBM traffic -> ~11us at 23.3 TB/s. Coordinates are
// computed once per pixel (V_WMMA_F32_16X16X4_F32 performs the reference's
// 3x3 einsum for 16 pixels per instruction) and reused across 64 channels.

typedef __attribute__((ext_vector_type(2))) float v2f;
typedef __attribute__((ext_vector_type(8))) float v8f;

#define BEV_B  8
#define BEV_C  64
#define BEV_H  192
#define BEV_W  320
#define BEV_HW (BEV_H * BEV_W)
#define EPS_D  1e-16f

__global__ __launch_bounds__(256) void bev_main_kernel(
    const float* __restrict__ x,
    const float* __restrict__ h_cam, const float* __restrict__ p_cam,
    const float* __restrict__ fu,    const float* __restrict__ fv,
    const int*   __restrict__ plane_h,
    float* __restrict__ out)
{
    const int b    = blockIdx.y;                          // batch (uniform per block)
    const int n    = blockIdx.x * blockDim.x + threadIdx.x; // pixel in [0, HW)
    const int lane = threadIdx.x & 31;
    const int n0   = n - lane;                            // wave-base pixel

    // ---- per-batch scalars (uniform -> scalar loads) ----
    const float h   = h_cam[b];
    const float p   = p_cam[b];
    const float fuu = fu[b];
    const float fvv = fv[b];
    const float ph  = (float)plane_h[0];

    const float sp = sinf(p), cp = cosf(p);
    const float h_ref = h - ph;
    const float cu = BEV_W * 0.5f, cv = BEV_H * 0.5f;

    // mats (3x3), rows:
    const float m00 = cu * cp,            m01 = -fuu, m02 = cu * h_ref * sp;
    const float m10 = cv * cp - fvv * sp,             m12 = h_ref * (fvv * cp + cv * sp);
    const float m20 = cp,                              m22 = h_ref * sp;
    const float s   = h / (sp * (cv * cp + fvv * sp) + EPS_D);
    const float cx0 = h * cp / (sp + EPS_D);           // centers[b][0]; centers[b][1]=0

    // ---- A matrix (16x4 f32): lane m<16 holds row m, K=0/1 in VGPR0/1;
    //      lanes 16-31 hold K=2/3. Rows 3..15 are zero. ----
    const int r = lane & 15;
    const float c0 = (r == 0) ? m00 : (r == 1) ? m10 : (r == 2) ? m20 : 0.0f;
    const float c1 = (r == 0) ? m01 : 0.0f;            // m11 = m21 = 0
    const float c2 = (r == 0) ? m02 : (r == 1) ? m12 : (r == 2) ? m22 : 0.0f;
    v2f A;
    A.x = (lane < 16) ? c0 : c2;   // K=0 | K=2
    A.y = (lane < 16) ? c1 : 0.0f; // K=1 | K=3

    // ---- B matrix (4x16 f32) for 16 pixels starting at `base`:
    //      row0=wx, row1=wy, row2=1, row3=0 ----
    auto make_world = [&](int base) -> v2f {
        const int px = base + (lane & 15);
        const int vv = px / BEV_W;
        const int uu = px - vv * BEV_W;
        const float wx = cx0 + s * (cv - (float)vv);
        const float wy =       s * (cu - (float)uu);
        v2f Bm;
        Bm.x = (lane < 16) ? wx : 1.0f; // K=0 | K=2
        Bm.y = (lane < 16) ? wy : 0.0f; // K=1 | K=3
        return Bm;
    };

    const v2f B0 = make_world(n0);
    const v2f B1 = make_world(n0 + 16);
    v8f Z = {};

    // D[m][nn] = sum_k mats[m][k] * world[k][nn]  ==  uvw[m][pixel nn]
    // (args: neg_a, A, neg_b, B, c_mod, C, reuse_a, reuse_b)
    v8f D0 = __builtin_amdgcn_wmma_f32_16x16x4_f32(false, A, false, B0, (short)0, Z, false, false);
    v8f D1 = __builtin_amdgcn_wmma_f32_16x16x4_f32(false, A, false, B1, (short)0, Z, false, false);

    // C/D layout: VGPR j, lanes 0-15 -> row M=j, col N=lane. So for lane p<16,
    // D0[0..2] are uvw of pixel n0+p. Move D1's lanes 0-15 into lanes 16-31.
    const int  sl = lane & 15;
    const float e0 = __shfl(D1[0], sl);
    const float e1 = __shfl(D1[1], sl);
    const float e2 = __shfl(D1[2], sl);
    const float uvw0 = (lane < 16) ? D0[0] : e0;
    const float uvw1 = (lane < 16) ? D0[1] : e1;
    const float uvw2 = (lane < 16) ? D0[2] : e2;

    const float su = uvw0 / uvw2;
    const float sv = uvw1 / uvw2;

    const int fsu = (int)floorf(su);
    const int fsv = (int)floorf(sv);
    const int u0 = min(max(fsu,     0), BEV_W - 1);
    const int u1 = min(max(fsu + 1, 0), BEV_W - 1);
    const int v0 = min(max(fsv,     0), BEV_H - 1);
    const int v1 = min(max(fsv + 1, 0), BEV_H - 1);

    const float wa = ((float)v1 - sv) * ((float)u1 - su);
    const float wb = (sv - (float)v0) * ((float)u1 - su);
    const float wc = ((float)v1 - sv) * (su - (float)u0);
    const float wd = (sv - (float)v0) * (su - (float)u0);

    const int i00 = v0 * BEV_W + u0;
    const int i10 = v1 * BEV_W + u0;
    const int i01 = v0 * BEV_W + u1;
    const int i11 = v1 * BEV_W + u1;

    // ---- channel loop: 4 gathers + 4 FMA + 1 store per channel (amortizes
    //      all coordinate math over C=64; x fits in 192MB L2) ----
    const float* xb = x   + (size_t)b * BEV_C * BEV_HW;
    float*       ob = out + (size_t)b * BEV_C * BEV_HW + n;
#pragma unroll 4
    for (int c = 0; c < BEV_C; ++c) {
        const float a00 = xb[i00];
        const float a10 = xb[i10];
        const float a01 = xb[i01];
        const float a11 = xb[i11];
        ob[(size_t)c * BEV_HW] = wa * a00 + wb * a10 + wc * a01 + wd * a11;
        xb += BEV_HW;
    }
}

// Tiny kernel: scales[B] and centers[B,3] appended after `out` in d_out.
__global__ void bev_params_kernel(
    const float* __restrict__ h_cam, const float* __restrict__ p_cam,
    const float* __restrict__ fv, float* __restrict__ out)
{
    const int b = threadIdx.x;
    if (b >= BEV_B) return;
    const float h  = h_cam[b];
    const float p  = p_cam[b];
    const float fvv = fv[b];
    const float sp = sinf(p), cp = cosf(p);
    const float cv = BEV_H * 0.5f;

    float* scales  = out + (size_t)BEV_B * BEV_C * BEV_HW;
    float* centers = scales + BEV_B;
    scales[b]          = h / (sp * (cv * cp + fvv * sp) + EPS_D);
    centers[b * 3 + 0] = h * cp / (sp + EPS_D);
    centers[b * 3 + 1] = 0.0f;
    centers[b * 3 + 2] = 1.0f;
}

extern "C" void kernel_launch(void* const* d_in, const int* in_sizes, int n_in,
                              void* d_out, int out_size, void* d_ws, size_t ws_size,
                              hipStream_t stream) {
    const float* x     = (const float*)d_in[0];
    const float* h_cam = (const float*)d_in[1];
    const float* p_cam = (const float*)d_in[2];
    const float* fu    = (const float*)d_in[3];
    const float* fv    = (const float*)d_in[4];
    const int*   ph    = (const int*)  d_in[5];
    float* out = (float*)d_out;

    dim3 grid(BEV_HW / 256, BEV_B); // 61440/256 = 240 exact, 8 waves/block
    bev_main_kernel<<<grid, 256, 0, stream>>>(x, h_cam, p_cam, fu, fv, ph, out);
    bev_params_kernel<<<1, 32, 0, stream>>>(h_cam, p_cam, fv, out);
}